// GCN_27865747817169
// MI455X (gfx1250) — compile-verified
//
#include <hip/hip_runtime.h>
#include <hip/hip_bf16.h>

typedef float v2f __attribute__((ext_vector_type(2)));
typedef float v8f __attribute__((ext_vector_type(8)));

#define LDS_STRIDE 260   // 256 + 4 pad -> bank = (4*row + k) % 64, conflict-free

__device__ __forceinline__ void atomAddF32(float* p, float v) {
    // hardware global_atomic_add_f32 (no CAS loop)
    unsafeAtomicAdd(p, v);
}

// ---------------- degree / normalization ----------------

__global__ __launch_bounds__(256) void k_deg_init(float* __restrict__ deg, int n) {
    int i = blockIdx.x * 256 + threadIdx.x;
    if (i < n) deg[i] = 1.0f;               // self loop
}

__global__ __launch_bounds__(256) void k_deg_edges(const int* __restrict__ dst,
                                                   float* __restrict__ deg, int e) {
    int i = blockIdx.x * 256 + threadIdx.x;
    if (i < e) atomAddF32(&deg[dst[i]], 1.0f);
}

__global__ __launch_bounds__(256) void k_rsqrt(float* __restrict__ deg, int n) {
    int i = blockIdx.x * 256 + threadIdx.x;
    if (i < n) deg[i] = __frsqrt_rn(deg[i]);   // deg -> dis = deg^{-1/2}
}

// ---------------- GEMM1: h0 = x[N,256] @ W1[256,128] (f32 WMMA) ----------------
// block = 16 rows of x in LDS; 8 waves -> 8 column tiles of 16.

__global__ __launch_bounds__(256) void k_gemm1(const float* __restrict__ x,
                                               const float* __restrict__ W1,
                                               float* __restrict__ h0) {
    __shared__ float As[16 * LDS_STRIDE];
    const int tid = threadIdx.x;
    const int r0  = blockIdx.x * 16;

    // cooperative, fully-coalesced float4 fill of the 16x256 strip
    const float4* xv = (const float4*)(x + (size_t)r0 * 256);
    #pragma unroll
    for (int i = 0; i < 4; ++i) {
        int j   = i * 256 + tid;      // float4 index 0..1023
        int row = j >> 6;             // 64 float4 per row
        int c4  = j & 63;
        float4 v = xv[(size_t)row * 64 + c4];
        float* d = &As[row * LDS_STRIDE + c4 * 4];
        d[0] = v.x; d[1] = v.y; d[2] = v.z; d[3] = v.w;
    }
    __syncthreads();

    const int wave  = tid >> 5;         // 0..7 -> column tile
    const int lane  = tid & 31;
    const int half  = lane >> 4;        // 0: K=k0,k0+1   1: K=k0+2,k0+3
    const int l16   = lane & 15;
    const int c0    = wave * 16;
    const int kboff = half * 2;

    v8f acc = {0.f, 0.f, 0.f, 0.f, 0.f, 0.f, 0.f, 0.f};
    for (int k0 = 0; k0 < 256; k0 += 4) {
        v2f a, b;
        const float* ap = &As[l16 * LDS_STRIDE + k0 + kboff];
        a.x = ap[0]; a.y = ap[1];
        const float* bp = &W1[(size_t)(k0 + kboff) * 128 + c0 + l16];
        b.x = bp[0]; b.y = bp[128];
        acc = __builtin_amdgcn_wmma_f32_16x16x4_f32(false, a, false, b,
                                                    (short)0, acc, false, false);
    }
    // C layout: VGPR i -> rows (i, i+8) across lane halves
    float* out = h0 + (size_t)(r0 + half * 8) * 128 + c0 + l16;
    #pragma unroll
    for (int i = 0; i < 8; ++i) out[(size_t)i * 128] = acc[i];
}

// ---------------- layer-1 self loop + bias, scatter-add ----------------

__global__ __launch_bounds__(256) void k_init_out1(const float* __restrict__ h0,
                                                   const float* __restrict__ b1,
                                                   const float* __restrict__ dis,
                                                   float* __restrict__ out1, int n) {
    int g = blockIdx.x * 256 + threadIdx.x;        // over n*32 float4s
    if (g >= n * 32) return;
    int node = g >> 5;
    int f4   = g & 31;
    float d2 = dis[node] * dis[node];              // self-loop norm = 1/deg
    float4 v = ((const float4*)h0)[(size_t)g];
    float4 bb = ((const float4*)b1)[f4];
    float4 o;
    o.x = bb.x + v.x * d2; o.y = bb.y + v.y * d2;
    o.z = bb.z + v.z * d2; o.w = bb.w + v.w * d2;
    ((float4*)out1)[(size_t)g] = o;
}

__global__ __launch_bounds__(256) void k_scatter128(const int* __restrict__ src,
                                                    const int* __restrict__ dst,
                                                    const float* __restrict__ dis,
                                                    const float* __restrict__ h,
                                                    float* __restrict__ out, int e) {
    int g = blockIdx.x * 256 + threadIdx.x;        // 1 wave per edge
    int edge = g >> 5;
    if (edge >= e) return;
    int lane = threadIdx.x & 31;
    int s = src[edge], d = dst[edge];
    float nrm = dis[s] * dis[d];
    float4 v = *(const float4*)(h + (size_t)s * 128 + lane * 4);
    float* o = out + (size_t)d * 128 + lane * 4;
    atomAddF32(o + 0, v.x * nrm);
    atomAddF32(o + 1, v.y * nrm);
    atomAddF32(o + 2, v.z * nrm);
    atomAddF32(o + 3, v.w * nrm);
}

// ---------------- GEMM2: h2 = relu(out1)[N,128] @ W2[128,16] ----------------

__global__ __launch_bounds__(256) void k_gemm2(const float* __restrict__ out1,
                                               const float* __restrict__ W2,
                                               float* __restrict__ h2, int ntiles) {
    int wg = (blockIdx.x * 256 + threadIdx.x) >> 5;  // one wave = one 16-row tile
    if (wg >= ntiles) return;
    const int lane  = threadIdx.x & 31;
    const int half  = lane >> 4;
    const int l16   = lane & 15;
    const int kboff = half * 2;
    const int r0    = wg * 16;

    const float* arow = out1 + (size_t)(r0 + l16) * 128;
    v8f acc = {0.f, 0.f, 0.f, 0.f, 0.f, 0.f, 0.f, 0.f};
    for (int k0 = 0; k0 < 128; k0 += 4) {
        v2f a, b;
        a.x = fmaxf(arow[k0 + kboff], 0.f);          // fused ReLU
        a.y = fmaxf(arow[k0 + kboff + 1], 0.f);
        const float* bp = &W2[(size_t)(k0 + kboff) * 16 + l16];
        b.x = bp[0]; b.y = bp[16];
        acc = __builtin_amdgcn_wmma_f32_16x16x4_f32(false, a, false, b,
                                                    (short)0, acc, false, false);
    }
    float* out = h2 + (size_t)(r0 + half * 8) * 16 + l16;
    #pragma unroll
    for (int i = 0; i < 8; ++i) out[(size_t)i * 16] = acc[i];
}

// ---------------- layer-2 init + scatter + softmax ----------------

__global__ __launch_bounds__(256) void k_init_out2(const float* __restrict__ h2,
                                                   const float* __restrict__ b2,
                                                   const float* __restrict__ dis,
                                                   float* __restrict__ logits, int n) {
    int g = blockIdx.x * 256 + threadIdx.x;        // over n*16
    if (g >= n * 16) return;
    int node = g >> 4;
    int f    = g & 15;
    float d2 = dis[node] * dis[node];
    logits[g] = b2[f] + h2[g] * d2;
}

__global__ __launch_bounds__(256) void k_scatter16(const int* __restrict__ src,
                                                   const int* __restrict__ dst,
                                                   const float* __restrict__ dis,
                                                   const float* __restrict__ h2,
                                                   float* __restrict__ logits, int e) {
    int g = blockIdx.x * 256 + threadIdx.x;        // half-wave (16 lanes) per edge
    int edge = g >> 4;
    if (edge >= e) return;
    int f = g & 15;
    int s = src[edge], d = dst[edge];
    float nrm = dis[s] * dis[d];
    float v = h2[(size_t)s * 16 + f];
    atomAddF32(&logits[(size_t)d * 16 + f], v * nrm);
}

__global__ __launch_bounds__(256) void k_softmax16(const float* __restrict__ logits,
                                                   float* __restrict__ probs, int n) {
    int g = blockIdx.x * 256 + threadIdx.x;        // n*16 threads, exact multiple of 256
    if (g >= n * 16) return;
    float v = logits[g];
    float m = v;
    #pragma unroll
    for (int o = 8; o >= 1; o >>= 1) m = fmaxf(m, __shfl_xor(m, o, 16));
    float e = __expf(v - m);
    float s = e;
    #pragma unroll
    for (int o = 8; o >= 1; o >>= 1) s += __shfl_xor(s, o, 16);
    probs[g] = e / s;
}

// ---------------- host launcher ----------------

extern "C" void kernel_launch(void* const* d_in, const int* in_sizes, int n_in,
                              void* d_out, int out_size, void* d_ws, size_t ws_size,
                              hipStream_t stream) {
    const float* x  = (const float*)d_in[0];
    const int*   ei = (const int*)d_in[1];     // int32 per harness contract
    const float* W1 = (const float*)d_in[2];
    const float* b1 = (const float*)d_in[3];
    const float* W2 = (const float*)d_in[4];
    const float* b2 = (const float*)d_in[5];

    const int N = in_sizes[0] / 256;           // IN_FEAT = 256
    const int E = in_sizes[1] / 2;
    const int* src = ei;
    const int* dst = ei + E;

    // workspace layout
    char*  ws  = (char*)d_ws;
    size_t off = 0;
    auto alloc = [&](size_t bytes) -> void* {
        void* p = ws + off;
        off += (bytes + 255) & ~(size_t)255;
        return p;
    };
    float* dis  = (float*)alloc((size_t)N * 4);          // deg, then rsqrt(deg)
    float* h0   = (float*)alloc((size_t)N * 128 * 4);    // x @ W1
    float* out1 = (float*)alloc((size_t)N * 128 * 4);    // layer-1 aggregate (pre-relu)
    float* h2   = (float*)alloc((size_t)N * 16 * 4);     // relu(out1) @ W2
    float* logits = (float*)d_out;
    float* probs  = (float*)d_out + (size_t)N * 16;

    auto cdiv = [](long long a, long long b) { return (int)((a + b - 1) / b); };

    // 1) normalization coefficients
    k_deg_init <<<cdiv(N, 256), 256, 0, stream>>>(dis, N);
    k_deg_edges<<<cdiv(E, 256), 256, 0, stream>>>(dst, dis, E);
    k_rsqrt    <<<cdiv(N, 256), 256, 0, stream>>>(dis, N);

    // 2) layer 1
    k_gemm1    <<<N / 16, 256, 0, stream>>>(x, W1, h0);
    k_init_out1<<<cdiv((long long)N * 32, 256), 256, 0, stream>>>(h0, b1, dis, out1, N);
    k_scatter128<<<cdiv((long long)E * 32, 256), 256, 0, stream>>>(src, dst, dis, h0, out1, E);

    // 3) layer 2 (ReLU fused into GEMM2 loads)
    int ntiles = N / 16;
    k_gemm2    <<<cdiv(ntiles, 8), 256, 0, stream>>>(out1, W2, h2, ntiles);
    k_init_out2<<<cdiv((long long)N * 16, 256), 256, 0, stream>>>(h2, b2, dis, logits, N);
    k_scatter16<<<cdiv((long long)E * 16, 256), 256, 0, stream>>>(src, dst, dis, h2, logits, E);

    // 4) softmax
    k_softmax16<<<cdiv((long long)N * 16, 256), 256, 0, stream>>>(logits, probs, N);
}